// Fields_47072841564978
// MI455X (gfx1250) — compile-verified
//
#include <hip/hip_runtime.h>

// Fast-math (nnan/nsz) so maxnum lowers to a single v_max_num without canonicalize.
#pragma float_control(precise, off)

// CDNA5 / gfx1250, wave32. Batched tiny-MLP (3->16->16->3) over 1M points x 16 fields,
// exact-fp32 V_WMMA_F32_16X16X4_F32 (D = A(16x4) * B(4x16) + C).
//
// Layouts per CDNA5 ISA (wave32):
//   A 16x4 f32 : lane l (l&15 = M). VGPR0 = K0 (lo half) / K2 (hi half); VGPR1 = K1 / K3.
//   B 4x16 f32 : lane l (l&15 = N). VGPR0 = K0 (lo half) / K2 (hi half); VGPR1 = K1 / K3.
//   C/D 16x16  : VGPR v = row M=v (lo half) and M=v+8 (hi half), col N = lane&15.
//
// Shuffle-free layer chaining: D vgpr pair (2c, 2c+1) is used directly as the B operand of
// K-chunk c. It presents hidden unit p = 2c + j + 8*half at K-position q = 4c + j + 2*half
// (both bijections over 0..15); we compensate by loading the NEXT layer's weight columns in
// p-order at q-positions: a[c][r] = W[m][2c + r + (hi?8:0)]. Hidden-unit numbering is
// internal, so this is exact — zero cross-lane data movement between layers.
//
// Zero-column trick: A1's K=3 column is all zeros, so B's K=3 slot (bx[1], hi half) may hold
// any finite value — no cndmask needed on the x row-1 load.
//
// ReLU uses __builtin_fmaxf (NOT fmaxf): the OCML fmax wrapper canonicalizes its inputs
// (v_max v,v,v), doubling ReLU cost; the builtin + fast-math pragma emits bare v_max_num.

typedef float v2f __attribute__((ext_vector_type(2)));
typedef float v8f __attribute__((ext_vector_type(8)));

#define WMMA4F32(A, B, C) \
  __builtin_amdgcn_wmma_f32_16x16x4_f32(false, (A), false, (B), (short)0, (C), false, false)

#define TILES_PER_WAVE 16
#define D_FIELDS 16

__device__ __forceinline__ v8f relu8(v8f h) {
#pragma unroll
  for (int v = 0; v < 8; ++v) h[v] = __builtin_fmaxf(h[v], 0.0f);
  return h;
}

__global__ __launch_bounds__(256) void fields_mlp_wmma(
    const float* __restrict__ x,
    const float* __restrict__ W1, const float* __restrict__ b1,
    const float* __restrict__ W2, const float* __restrict__ b2,
    const float* __restrict__ W3, const float* __restrict__ b3,
    float* __restrict__ out, int N, int groups) {
  const int tid  = blockIdx.x * blockDim.x + threadIdx.x;
  const int wave = tid >> 5;
  const int lane = threadIdx.x & 31;
  const int m    = lane & 15;          // M row (A/D) or N col (B/D)
  const bool hi  = (lane & 16) != 0;   // upper half-wave

  const int d     = wave & (D_FIELDS - 1);
  const int group = wave >> 4;
  if (group >= groups) return;

  // ---- Per-field constants hoisted into registers -------------------------
  // Layer 1 A: W1[d] is 16x3, K padded to 4 with zeros. W1 flat [D,16,3].
  v2f a1;
  a1[0] = hi ? W1[d * 48 + m * 3 + 2] : W1[d * 48 + m * 3 + 0];
  a1[1] = hi ? 0.0f : W1[d * 48 + m * 3 + 1];

  // Layer 2 A chunks with column order compensating the direct D->B feed.
  // W2 flat [D,16,16] (row = out channel, col = hidden-in).
  v2f a2[4];
#pragma unroll
  for (int c = 0; c < 4; ++c) {
    a2[c][0] = W2[d * 256 + m * 16 + 2 * c + 0 + (hi ? 8 : 0)];
    a2[c][1] = W2[d * 256 + m * 16 + 2 * c + 1 + (hi ? 8 : 0)];
  }

  // Layer 3 A chunks (rows >= 3 zero), same column compensation. W3 flat [D,3,16].
  v2f a3[4];
#pragma unroll
  for (int c = 0; c < 4; ++c) {
    a3[c][0] = (m < 3) ? W3[d * 48 + m * 16 + 2 * c + 0 + (hi ? 8 : 0)] : 0.0f;
    a3[c][1] = (m < 3) ? W3[d * 48 + m * 16 + 2 * c + 1 + (hi ? 8 : 0)] : 0.0f;
  }

  // Bias C matrices (broadcast along N). C/D vgpr v holds rows v (lo) / v+8 (hi).
  v8f c1v, c2v, c3v;
#pragma unroll
  for (int v = 0; v < 8; ++v) {
    c1v[v] = b1[d * 16 + v + (hi ? 8 : 0)];
    c2v[v] = b2[d * 16 + v + (hi ? 8 : 0)];
    c3v[v] = (!hi && v < 3) ? b3[d * 3 + v] : 0.0f;
  }

  const int tilesTotal = (N + 15) >> 4;
  const int t0 = group * TILES_PER_WAVE;
  const int t1 = min(t0 + TILES_PER_WAVE, tilesTotal);
  const int tFull = min(t1, N >> 4);   // tiles with all 16 points in-bounds

  // 32-bit per-lane element offsets against uniform SGPR bases (max 48M elems -> fits).
  int offA = (hi ? 2 * N : 0) + (t0 << 4) + m;   // x row 0 (lo) / row 2 (hi)
  int offB = N + (t0 << 4) + m;                  // x row 1
  int offO = d * 3 * N + (t0 << 4) + m;          // out row 0 of field d
  const int lastA = (hi ? 2 * N : 0) + N - 1;    // clamp bounds for prefetch
  const int lastB = 2 * N - 1;

  // ---- Full tiles: 1-deep software pipeline on the x loads ----------------
  if (t0 < tFull) {
    float ld0 = x[offA];
    float ld1 = x[offB];
    for (int t = t0; t < tFull; ++t) {
      const float c0 = ld0, c1 = ld1;
      // Prefetch next tile's x (clamped; clamped garbage is never consumed).
      offA += 16; offB += 16;
      ld0 = x[min(offA, lastA)];
      ld1 = x[min(offB, lastB)];

      v2f bx; bx[0] = c0; bx[1] = c1;   // hi half K=3 slot multiplied by zero A column

      v8f h = relu8(WMMA4F32(a1, bx, c1v));

      v8f g = c2v;
#pragma unroll
      for (int c = 0; c < 4; ++c) {
        v2f b; b[0] = h[2 * c]; b[1] = h[2 * c + 1];
        g = WMMA4F32(a2[c], b, g);
      }
      g = relu8(g);

      v8f o = c3v;
#pragma unroll
      for (int c = 0; c < 4; ++c) {
        v2f b; b[0] = g[2 * c]; b[1] = g[2 * c + 1];
        o = WMMA4F32(a3[c], b, o);
      }

      // Rows 0..2 live in D vgprs 0..2, lanes 0..15. out flat [D, 3, N].
      if (!hi) {
        out[offO]         = o[0];
        out[offO + N]     = o[1];
        out[offO + 2 * N] = o[2];
      }
      offO += 16;
    }
  }

  // ---- Tail tile (N % 16 != 0 only; never taken for N = 1e6) --------------
  const size_t Ns = (size_t)N;
  for (int t = tFull; t < t1; ++t) {
    const int n  = (t << 4) + m;
    const int nc = (n < N) ? n : (N - 1);
    v2f bx;
    bx[0] = hi ? x[2 * Ns + nc] : x[nc];
    bx[1] = x[Ns + nc];

    v8f h = relu8(WMMA4F32(a1, bx, c1v));
    v8f g = c2v;
#pragma unroll
    for (int c = 0; c < 4; ++c) {
      v2f b; b[0] = h[2 * c]; b[1] = h[2 * c + 1];
      g = WMMA4F32(a2[c], b, g);
    }
    g = relu8(g);
    v8f o = c3v;
#pragma unroll
    for (int c = 0; c < 4; ++c) {
      v2f b; b[0] = g[2 * c]; b[1] = g[2 * c + 1];
      o = WMMA4F32(a3[c], b, o);
    }
    if (!hi && n < N) {
      float* op = out + (size_t)d * 3 * Ns + n;
      op[0]      = o[0];
      op[Ns]     = o[1];
      op[2 * Ns] = o[2];
    }
  }
}

extern "C" void kernel_launch(void* const* d_in, const int* in_sizes, int n_in,
                              void* d_out, int out_size, void* d_ws, size_t ws_size,
                              hipStream_t stream) {
  (void)n_in; (void)out_size; (void)d_ws; (void)ws_size;
  const float* x  = (const float*)d_in[0];
  const float* W1 = (const float*)d_in[1];
  const float* b1 = (const float*)d_in[2];
  const float* W2 = (const float*)d_in[3];
  const float* b2 = (const float*)d_in[4];
  const float* W3 = (const float*)d_in[5];
  const float* b3 = (const float*)d_in[6];
  // d_in[7] is the python scalar D (=16); hardcoded as D_FIELDS.

  const int N      = in_sizes[0] / 3;
  const int tiles  = (N + 15) / 16;
  const int groups = (tiles + TILES_PER_WAVE - 1) / TILES_PER_WAVE;
  const long long waves   = (long long)groups * D_FIELDS;
  const long long threads = waves * 32;
  const int block = 256;
  const int grid  = (int)((threads + block - 1) / block);

  fields_mlp_wmma<<<grid, block, 0, stream>>>(x, W1, b1, W2, b2, W3, b3,
                                              (float*)d_out, N, groups);
}